// PitWrapper_40596030881827
// MI455X (gfx1250) — compile-verified
//
#include <hip/hip_runtime.h>

typedef __attribute__((ext_vector_type(2))) float v2f;
typedef __attribute__((ext_vector_type(8))) float v8f;
typedef __attribute__((ext_vector_type(4))) float f4v;

#define BATCH    8
#define KTOT     (4000 * 257)          // positions (T*F) per batch = 1,028,000
#define TILE     2048                  // positions staged per workgroup tile
#define NTILES   ((KTOT + TILE - 1) / TILE)  // 502
#define NBLK     64                    // workgroups per batch
#define NTHREADS 256
#define WAVES    8
#define POSW     (TILE / WAVES)        // 256 positions per wave per tile
#define ITERS    (POSW / 8)            // 32 WMMAs per wave per tile (8 pos each)
#define TOFF     (TILE * 3 + 24)       // T-region float offset (24-float pad: bank-conflict-free)
#define LDSF     (TOFF + TILE * 3)     // 12312 floats staging region

__global__ __launch_bounds__(NTHREADS) void pit_partial(
    const float* __restrict__ preds,
    const float* __restrict__ targets,
    float* __restrict__ partials) {
  __shared__ __align__(16) float lds[LDSF];
  __shared__ float dmat[WAVES * 256];  // per-wave 16x16 D dump

  const int b   = blockIdx.x / NBLK;
  const int blk = blockIdx.x % NBLK;
  const int tid = threadIdx.x;
  const int lane = tid & 31;
  const int w    = tid >> 5;

  // Per-lane LDS float index for the shared A/B fragment.
  // Rows 0-5 = stream1 v-vector, rows 6-11 = stream2 v-vector (interleaved 4-pos blocks).
  const int l16 = lane & 15;
  int baseIdx;
  if (l16 < 12) {
    const int g      = l16 / 6;           // stream group (0 or 1)
    const int m      = l16 % 6;           // vector entry 0..5
    const int chan   = (m >= 3) ? (m - 3) : m;
    const int isT    = (m >= 3);
    const int posOff = (lane >> 4) * 2;   // upper lane half holds K=2,3 (pos+2)
    baseIdx = (isT ? TOFF : 0) + (w * POSW + posOff) * 3 + g * 12 + chan;
  } else {
    baseIdx = 0;                          // rows/cols 12-15 of D are never read
  }

  v8f acc = {0.f, 0.f, 0.f, 0.f, 0.f, 0.f, 0.f, 0.f};

  const float* psrc = preds   + (size_t)b * KTOT * 3;
  const float* tsrc = targets + (size_t)b * KTOT * 3;

  for (int tile = blk; tile < NTILES; tile += NBLK) {
    const int pos0 = tile * TILE;
    const int rem  = KTOT - pos0;
    const int validF4 = (rem >= TILE) ? (TILE * 3 / 4) : (rem * 3 / 4);

    __syncthreads();  // previous tile's compute done before overwriting LDS

    // Stage P and T slices with coalesced 128-bit loads; zero-pad the tail
    // (zeros contribute nothing to the Gram sums).
    const f4v* ps = (const f4v*)(psrc + (size_t)pos0 * 3);
    const f4v* ts = (const f4v*)(tsrc + (size_t)pos0 * 3);
    f4v* ldsP = (f4v*)lds;
    f4v* ldsT = (f4v*)(lds + TOFF);
    for (int idx = tid; idx < TILE * 3 / 4; idx += NTHREADS) {
      f4v vp = {0.f, 0.f, 0.f, 0.f};
      f4v vt = {0.f, 0.f, 0.f, 0.f};
      if (idx < validF4) { vp = ps[idx]; vt = ts[idx]; }
      ldsP[idx] = vp;
      ldsT[idx] = vt;
    }
    __syncthreads();

    // Chained f32 WMMA rank-4 updates: D += A x A (A layout == B layout here).
#pragma unroll
    for (int i = 0; i < ITERS; ++i) {
      const float a0 = lds[baseIdx + i * 24];       // K = 0 / 2
      const float a1 = lds[baseIdx + i * 24 + 3];   // K = 1 / 3
      v2f A;
      A.x = a0;
      A.y = a1;
      acc = __builtin_amdgcn_wmma_f32_16x16x4_f32(
          false, A, false, A, (short)0, acc, false, false);
    }
  }

  // Dump per-wave D (16x16) to LDS: VGPR r, lanes 0-15 -> M=r; lanes 16-31 -> M=r+8.
  {
    const int half = lane >> 4;
    const int n    = lane & 15;
#pragma unroll
    for (int r = 0; r < 8; ++r) {
      dmat[w * 256 + (r + 8 * half) * 16 + n] = acc[r];
    }
  }
  __syncthreads();

  // Fold the two diagonal 6x6 blocks and the 8 waves -> 36 partial sums.
  if (tid < 36) {
    const int mi = tid / 6, ni = tid % 6;
    float s = 0.f;
    for (int ww = 0; ww < WAVES; ++ww) {
      const float* Dw = dmat + ww * 256;
      s += Dw[mi * 16 + ni] + Dw[(6 + mi) * 16 + (6 + ni)];
    }
    partials[(size_t)(b * NBLK + blk) * 36 + tid] = s;
  }
}

__global__ void pit_final(const float* __restrict__ partials,
                          float* __restrict__ out) {
  __shared__ float S6[36];
  const int b = blockIdx.x;
  const int tid = threadIdx.x;
  if (tid < 36) {
    float s = 0.f;
    for (int j = 0; j < NBLK; ++j)
      s += partials[(size_t)(b * NBLK + j) * 36 + tid];
    S6[tid] = s;
  }
  __syncthreads();
  if (tid == 0) {
    const float inv = 1.0f / (float)KTOT;
    float lm[3][3];
    for (int i = 0; i < 3; ++i)
      for (int j = 0; j < 3; ++j)
        lm[i][j] = (S6[i * 6 + i] - 2.f * S6[i * 6 + 3 + j] +
                    S6[(3 + j) * 6 + 3 + j]) * inv;
    const int perms[6][3] = {{0,1,2},{0,2,1},{1,0,2},{1,2,0},{2,0,1},{2,1,0}};
    float best = 3.402823466e38f;
    int bi = 0;
    for (int p = 0; p < 6; ++p) {
      const float c = (lm[0][perms[p][0]] + lm[1][perms[p][1]] +
                       lm[2][perms[p][2]]) * (1.f / 3.f);
      if (c < best) { best = c; bi = p; }   // strict <  == argmin first-min
    }
    out[b] = best;
    out[BATCH + b * 3 + 0] = (float)perms[bi][0];
    out[BATCH + b * 3 + 1] = (float)perms[bi][1];
    out[BATCH + b * 3 + 2] = (float)perms[bi][2];
  }
}

extern "C" void kernel_launch(void* const* d_in, const int* in_sizes, int n_in,
                              void* d_out, int out_size, void* d_ws, size_t ws_size,
                              hipStream_t stream) {
  const float* preds   = (const float*)d_in[0];
  const float* targets = (const float*)d_in[1];
  float* partials = (float*)d_ws;                 // 8*64*36 floats = 73,728 B
  float* out      = (float*)d_out;                // 8 loss + 24 perm values

  pit_partial<<<dim3(BATCH * NBLK), dim3(NTHREADS), 0, stream>>>(preds, targets, partials);
  pit_final<<<dim3(BATCH), dim3(64), 0, stream>>>(partials, out);
}